// HEARConv_47236050321666
// MI455X (gfx1250) — compile-verified
//
#include <hip/hip_runtime.h>
#include <hip/hip_bf16.h>

// HEARConv-style relational graph attention for MI455X (gfx1250, wave32).
// Pipeline: (1) per-relation node projections via V_WMMA_F32_16X16X4_F32 with
// W[r] staged K-pair-interleaved in LDS (one ds_load_b64 per B fragment,
// bank-conflict-free via 160-float row padding; bias folded in), (2) per-edge
// score + segment-max via encoded-uint atomic max, (3) exp + segment-sum
// denominator via f32 atomics, (4) scatter-add of a * feat[src] into (N,H,D).
// Whole pipeline is HBM-bound (~1.2 GB moved vs 6.5 GFLOP) => fp32 WMMA keeps
// reference numerics at zero cost.

typedef float v2f __attribute__((ext_vector_type(2)));
typedef float v8f __attribute__((ext_vector_type(8)));

#define D_FEAT 64     // in_feats
#define HK     64     // H*K
#define NHEAD  2
#define KATT   32
#define NEG_SLOPE 0.2f
#define TILES_PER_BLOCK 4   // 4 waves/block, one 16-row node tile per wave
#define KP_STRIDE 160       // floats per K-pair row in LDS (mod 64 == 32
                            // => odd kp rows start 32 banks later: the two
                            // half-wave groups of a b64 load cover banks
                            // 0..31 and 32..63 -> conflict-free)

// ---------------------------------------------------------------------------
// Phase 1: P[n, r, :] = X[n, :] @ W[r] + bias[r]      (X: (N,64), W: (R,64,64))
// Block = 128 threads = 4 waves sharing one relation's W tile in LDS.
// LDS layout: WP[kp][col] = (W[2kp][col], W[2kp+1][col]) as an 8-byte pair,
// so each WMMA B fragment is one aligned ds_load_b64 (no VGPR repacking).
// Each wave computes a 16x64 tile: 16 K-steps x 4 col chunks
// => 64 x v_wmma_f32_16x16x4_f32 per wave.
// ---------------------------------------------------------------------------
__global__ __launch_bounds__(32 * TILES_PER_BLOCK) void proj_kernel(
    const float* __restrict__ X,
    const float* __restrict__ W,
    const float* __restrict__ bias,
    float* __restrict__ P,
    int N, int R)
{
  __shared__ float Wlds[32 * KP_STRIDE];       // 20 KB of the 320 KB WGP LDS

  const int r = blockIdx.y;

  // Cooperative stage of W[r], K-pair interleaved with padding.
  // 1024 float4 global reads / 128 threads = 8 iters x (b128 + 4 ds_store).
  {
    const float4* __restrict__ Wg = (const float4*)(W + (size_t)r * D_FEAT * HK);
    for (int i = threadIdx.x; i < (D_FEAT * HK) / 4; i += blockDim.x) {
      const int k  = i >> 4;           // W row 0..63
      const int c0 = (i & 15) * 4;     // col base
      const float4 w = Wg[i];
      float* dstp = &Wlds[(k >> 1) * KP_STRIDE + (k & 1)];
      dstp[(c0 + 0) * 2] = w.x;
      dstp[(c0 + 1) * 2] = w.y;
      dstp[(c0 + 2) * 2] = w.z;
      dstp[(c0 + 3) * 2] = w.w;
    }
  }
  __syncthreads();

  const int wv   = threadIdx.x >> 5;
  const int lane = threadIdx.x & 31;
  const int half = lane >> 4;     // 0: K pair {0,1}; 1: K pair {2,3}
  const int l    = lane & 15;

  // tile is wave-uniform: scalarize so bounds checks are s_cbranch, not exec.
  const int tile = __builtin_amdgcn_readfirstlane(blockIdx.x * TILES_PER_BLOCK + wv);
  if (tile * 16 >= N) return;                  // uniform wave exit

  const int row  = tile * 16 + l;              // A-matrix M index (both halves)
  const int rowc = row < N ? row : (N - 1);    // clamp loads for ragged tile
  const float* __restrict__ Xr = X + (size_t)rowc * D_FEAT;

  v8f acc[4] = {v8f{}, v8f{}, v8f{}, v8f{}};

#pragma unroll 4
  for (int s = 0; s < 16; ++s) {               // K in steps of 4
    const int kp = 2 * s + half;               // K-pair index 0..31
    v2f a;
    a.x = Xr[2 * kp];                          // one global_load_b64
    a.y = Xr[2 * kp + 1];
    const float2* __restrict__ Wp = (const float2*)&Wlds[kp * KP_STRIDE];
#pragma unroll
    for (int c = 0; c < 4; ++c) {              // N in chunks of 16
      const float2 t = Wp[c * 16 + l];         // one ds_load_b64, no conflicts
      v2f b;
      b.x = t.x;
      b.y = t.y;
      acc[c] = __builtin_amdgcn_wmma_f32_16x16x4_f32(
          /*neg_a=*/false, a, /*neg_b=*/false, b,
          /*c_mod=*/(short)0, acc[c], /*reuse_a=*/false, /*reuse_b=*/false);
    }
  }

  // D layout: VGPR v -> row (v + 8*half), lane l -> column (c*16 + l).
  // Fast path (uniform): tile fully in-bounds -> straight coalesced stores.
  float* __restrict__ Pt = P + ((size_t)(tile * 16) * R + r) * HK;
  const size_t rstride = (size_t)R * HK;
  if (tile * 16 + 16 <= N) {
#pragma unroll
    for (int c = 0; c < 4; ++c) {
      const int col = c * 16 + l;
      const float bv = bias[(size_t)r * HK + col];
#pragma unroll
      for (int v = 0; v < 8; ++v)
        Pt[(size_t)(v + 8 * half) * rstride + col] = acc[c][v] + bv;
    }
  } else {
#pragma unroll
    for (int c = 0; c < 4; ++c) {
      const int col = c * 16 + l;
      const float bv = bias[(size_t)r * HK + col];
#pragma unroll
      for (int v = 0; v < 8; ++v) {
        const int m = tile * 16 + v + 8 * half;
        if (m < N)
          P[((size_t)m * R + r) * HK + col] = acc[c][v] + bv;
      }
    }
  }
}

// Monotonic uint encoding of float for atomic segment-max.
__device__ __forceinline__ unsigned enc_f32(float f) {
  unsigned u = __float_as_uint(f);
  return (u & 0x80000000u) ? ~u : (u | 0x80000000u);
}
__device__ __forceinline__ float dec_f32(unsigned u) {
  unsigned b = (u & 0x80000000u) ? (u & 0x7FFFFFFFu) : ~u;
  return __uint_as_float(b);
}

// ---------------------------------------------------------------------------
// Phase 2: per-edge scores + segment max.
// scores[e,h] = sum_k lrelu(Ps[src,r,h,k] + Pq[nid,r,h,k]) * attn[r,h,k]
// ---------------------------------------------------------------------------
__global__ __launch_bounds__(256) void score_kernel(
    const float* __restrict__ Ps, const float* __restrict__ Pq,
    const float* __restrict__ attn,
    const int* __restrict__ src, const int* __restrict__ dst,
    const int* __restrict__ rt, const int* __restrict__ nid,
    float* __restrict__ scores, unsigned* __restrict__ smax_enc,
    int E, int R)
{
  const int e = blockIdx.x * blockDim.x + threadIdx.x;
  if (e >= E) return;
  const int s = src[e], q = nid[e], r = rt[e], d = dst[e];
  const float4* __restrict__ ps = (const float4*)(Ps + ((size_t)s * R + r) * HK);
  const float4* __restrict__ pq = (const float4*)(Pq + ((size_t)q * R + r) * HK);
  const float4* __restrict__ at = (const float4*)(attn + (size_t)r * HK);
  __builtin_prefetch(ps + 8, 0, 0);   // global_prefetch_b8: pull 2nd cacheline
  __builtin_prefetch(pq + 8, 0, 0);

  float sc0 = 0.f, sc1 = 0.f;
#pragma unroll
  for (int k4 = 0; k4 < HK / 4; ++k4) {
    float4 a = ps[k4], b = pq[k4], t = at[k4];
    float v0 = a.x + b.x, v1 = a.y + b.y, v2 = a.z + b.z, v3 = a.w + b.w;
    v0 = v0 > 0.f ? v0 : v0 * NEG_SLOPE;
    v1 = v1 > 0.f ? v1 : v1 * NEG_SLOPE;
    v2 = v2 > 0.f ? v2 : v2 * NEG_SLOPE;
    v3 = v3 > 0.f ? v3 : v3 * NEG_SLOPE;
    const float p = v0 * t.x + v1 * t.y + v2 * t.z + v3 * t.w;
    if (k4 < (KATT / 4)) sc0 += p; else sc1 += p;   // head = k4 >= 8
  }
  scores[(size_t)e * NHEAD + 0] = sc0;
  scores[(size_t)e * NHEAD + 1] = sc1;
  atomicMax(&smax_enc[(size_t)d * NHEAD + 0], enc_f32(sc0));
  atomicMax(&smax_enc[(size_t)d * NHEAD + 1], enc_f32(sc1));
}

// ---------------------------------------------------------------------------
// Phase 3: z = exp(score - smax[dst]); denom[dst] += z  (z overwrites scores)
// ---------------------------------------------------------------------------
__global__ __launch_bounds__(256) void softmax_z_kernel(
    const int* __restrict__ dst,
    const unsigned* __restrict__ smax_enc,
    float* __restrict__ scores, float* __restrict__ denom, int E)
{
  const int t = blockIdx.x * blockDim.x + threadIdx.x;
  const int e = t >> 1, h = t & 1;
  if (e >= E) return;
  const int d = dst[e];
  const float m = dec_f32(smax_enc[(size_t)d * NHEAD + h]);
  const float z = __expf(scores[(size_t)e * NHEAD + h] - m);
  scores[(size_t)e * NHEAD + h] = z;
  atomicAdd(&denom[(size_t)d * NHEAD + h], z);
}

// ---------------------------------------------------------------------------
// Phase 4: out[dst, h, :] += (z / denom[dst,h]) * feat[src, :]
// One wave (32 lanes) per edge; lane covers k and k+32 for both heads.
// ---------------------------------------------------------------------------
__global__ __launch_bounds__(256) void agg_kernel(
    const float* __restrict__ feat,
    const int* __restrict__ src, const int* __restrict__ dst,
    const float* __restrict__ z, const float* __restrict__ denom,
    float* __restrict__ out, int E)
{
  const int t = blockIdx.x * blockDim.x + threadIdx.x;
  const int e = t >> 5;
  const int lane = t & 31;
  if (e >= E) return;
  const int s = src[e], d = dst[e];
  const float a0 = z[(size_t)e * NHEAD + 0] / denom[(size_t)d * NHEAD + 0];
  const float a1 = z[(size_t)e * NHEAD + 1] / denom[(size_t)d * NHEAD + 1];
  const float f0 = feat[(size_t)s * D_FEAT + lane];
  const float f1 = feat[(size_t)s * D_FEAT + lane + 32];
  float* o0 = out + ((size_t)d * NHEAD + 0) * D_FEAT;
  float* o1 = out + ((size_t)d * NHEAD + 1) * D_FEAT;
  atomicAdd(o0 + lane,      f0 * a0);
  atomicAdd(o0 + lane + 32, f1 * a0);
  atomicAdd(o1 + lane,      f0 * a1);
  atomicAdd(o1 + lane + 32, f1 * a1);
}

extern "C" void kernel_launch(void* const* d_in, const int* in_sizes, int n_in,
                              void* d_out, int out_size, void* d_ws, size_t ws_size,
                              hipStream_t stream) {
  // Input order: feat, W_src, b_src, qual_table, W_qual, b_qual, attn,
  //              src, dst, r_type, nid
  const float* feat   = (const float*)d_in[0];
  const float* W_src  = (const float*)d_in[1];
  const float* b_src  = (const float*)d_in[2];
  const float* qual   = (const float*)d_in[3];
  const float* W_qual = (const float*)d_in[4];
  const float* b_qual = (const float*)d_in[5];
  const float* attn   = (const float*)d_in[6];
  const int*   src    = (const int*)d_in[7];
  const int*   dst    = (const int*)d_in[8];
  const int*   rt     = (const int*)d_in[9];
  const int*   nid    = (const int*)d_in[10];
  float*       out    = (float*)d_out;

  const int N = in_sizes[0] / D_FEAT;                // 50000
  const int R = in_sizes[1] / (D_FEAT * HK);         // 8
  const int E = in_sizes[7];                         // 800000

  // Workspace layout (floats):
  //  [0)                P_src  : N*R*64   (bias folded)
  //  [N*R*64)           P_qual : N*R*64
  //  [2*N*R*64)         z      : E*2      (scores, then exp-z in place)
  //  then smax_enc (N*2 uints), denom (N*2 floats)
  float* P1 = (float*)d_ws;
  float* P2 = P1 + (size_t)N * R * HK;
  float* Z  = P2 + (size_t)N * R * HK;
  unsigned* SM = (unsigned*)(Z + (size_t)E * NHEAD);
  float* DEN = (float*)(SM + (size_t)N * NHEAD);

  // Init accumulators (0-bytes == float 0.0 and the encoded-max identity).
  hipMemsetAsync(out, 0, (size_t)out_size * sizeof(float), stream);
  hipMemsetAsync(SM, 0, (size_t)N * NHEAD * sizeof(unsigned), stream);
  hipMemsetAsync(DEN, 0, (size_t)N * NHEAD * sizeof(float), stream);

  // Phase 1: relation projections (WMMA f32, W[r] staged in LDS).
  const int ntiles = (N + 15) / 16;
  dim3 pg((ntiles + TILES_PER_BLOCK - 1) / TILES_PER_BLOCK, R);
  proj_kernel<<<pg, 32 * TILES_PER_BLOCK, 0, stream>>>(feat, W_src, b_src, P1, N, R);
  proj_kernel<<<pg, 32 * TILES_PER_BLOCK, 0, stream>>>(qual, W_qual, b_qual, P2, N, R);

  // Phase 2: edge scores + segment max.
  score_kernel<<<(E + 255) / 256, 256, 0, stream>>>(
      P1, P2, attn, src, dst, rt, nid, Z, SM, E, R);

  // Phase 3: z and denominators.
  softmax_z_kernel<<<(E * NHEAD + 255) / 256, 256, 0, stream>>>(dst, SM, Z, DEN, E);

  // Phase 4: weighted message scatter.
  agg_kernel<<<(int)(((size_t)E * 32 + 255) / 256), 256, 0, stream>>>(
      feat, src, dst, Z, DEN, out, E);
}